// CtcScorer_65635690218257
// MI455X (gfx1250) — compile-verified
//
#include <hip/hip_runtime.h>
#include <stdint.h>

#define T_DIM   4096
#define V_DIM   32000
#define NB_DIM  2048
#define START_T 11      // U-1 with U=12
#define TCHUNKS 8
#define TCH_LEN 512     // 4096 / 8
#define NEG_INF_F (-1e30f)
#define EOS_ID  1

#define K1_THREADS 512
#define K1_WAVES   (K1_THREADS / 32)
#define CHUNKS     (V_DIM / 4)           // 8000 b128 transfers per row
#define FULL_ITERS (CHUNKS / K1_THREADS) // 15 uniform iterations
#define TAIL_BASE  (FULL_ITERS * K1_THREADS)

// ---- CDNA5 async global->LDS DMA (ASYNCcnt path, verified to assemble) ---
__device__ __forceinline__ void async_copy_b128_to_lds(void* lds_dst, const void* gsrc) {
  uint32_t la = (uint32_t)(uintptr_t)lds_dst;   // low 32 bits of flat addr = LDS byte address
  uint64_t ga = (uint64_t)(uintptr_t)gsrc;
  asm volatile("global_load_async_to_lds_b128 %0, %1, off"
               :: "v"(la), "v"(ga)
               : "memory");
}

__device__ __forceinline__ void wait_asynccnt0() {
#if __has_builtin(__builtin_amdgcn_s_wait_asynccnt)
  __builtin_amdgcn_s_wait_asynccnt(0);
#else
  asm volatile("s_wait_asynccnt 0" ::: "memory");
#endif
}

// ---- K1: per-row logsumexp stats + fused gather, row staged in LDS -------
__global__ void __launch_bounds__(K1_THREADS) k1_row_lse_gather(
    const float* __restrict__ ctc_prob, const int* __restrict__ c,
    float* __restrict__ blank_lp, float* __restrict__ pcg)
{
  extern __shared__ float smem[];
  float* row = smem;               // V_DIM floats (128 KB)
  float* red = smem + V_DIM;       // reduction scratch (>= 34 floats)
  const int tid  = threadIdx.x;
  const int t    = blockIdx.x;
  const int lane = tid & 31, wid = tid >> 5;
  const float* g = ctc_prob + (size_t)t * V_DIM;

  // Stage the whole 128 KB row into LDS via async DMA.
  // 15 full-EXEC iterations + one divergent tail (320 chunks).
  for (int it = 0; it < FULL_ITERS; ++it) {
    int k = it * K1_THREADS + tid;
    async_copy_b128_to_lds(row + 4 * k, g + 4 * k);
  }
  {
    int k = TAIL_BASE + tid;
    if (k < CHUNKS) async_copy_b128_to_lds(row + 4 * k, g + 4 * k);
  }
  wait_asynccnt0();
  __syncthreads();

  const float4* row4 = reinterpret_cast<const float4*>(row);

  // Pass A: row max (wave32 shuffle reduction, then cross-wave via LDS).
  float m = NEG_INF_F;
  for (int k = tid; k < CHUNKS; k += K1_THREADS) {
    float4 v = row4[k];
    m = fmaxf(m, fmaxf(fmaxf(v.x, v.y), fmaxf(v.z, v.w)));
  }
  for (int o = 16; o; o >>= 1) m = fmaxf(m, __shfl_xor(m, o));
  if (lane == 0) red[wid] = m;
  __syncthreads();
  if (tid == 0) {
    float mm = red[0];
    for (int w = 1; w < K1_WAVES; ++w) mm = fmaxf(mm, red[w]);
    red[32] = mm;
  }
  __syncthreads();
  const float rowmax = red[32];

  // Pass B: sum of exp(x - max).
  float s = 0.f;
  for (int k = tid; k < CHUNKS; k += K1_THREADS) {
    float4 v = row4[k];
    s += __expf(v.x - rowmax) + __expf(v.y - rowmax)
       + __expf(v.z - rowmax) + __expf(v.w - rowmax);
  }
  for (int o = 16; o; o >>= 1) s += __shfl_xor(s, o);
  if (lane == 0) red[16 + wid] = s;
  __syncthreads();
  if (tid == 0) {
    float ss = 0.f;
    for (int w = 0; w < K1_WAVES; ++w) ss += red[16 + w];
    red[33] = rowmax + __logf(ss);   // lse[t]
  }
  __syncthreads();
  const float lse = red[33];

  if (tid == 0) blank_lp[t] = row[V_DIM - 1] - lse;

  // Fused gather of the 2048 candidate columns straight out of LDS
  // (avoids a second ~450 MB pass of scattered HBM/L2 reads).
  float* out = pcg + (size_t)t * NB_DIM;
  for (int j = tid; j < NB_DIM; j += K1_THREADS)
    out[j] = row[c[j]] - lse;
}

// ---- K2: cumulative sum of blank_lp over T (single block) ----------------
__global__ void __launch_bounds__(256) k2_cumsum(
    const float* __restrict__ blank_lp, float* __restrict__ cb)
{
  __shared__ float part[256];
  const int tid  = threadIdx.x;
  const int base = tid * 16;
  float run = 0.f, loc[16];
  for (int i = 0; i < 16; ++i) { run += blank_lp[base + i]; loc[i] = run; }
  part[tid] = run;
  __syncthreads();
  for (int off = 1; off < 256; off <<= 1) {
    float v = (tid >= off) ? part[tid - off] : 0.f;
    __syncthreads();
    if (tid >= off) part[tid] += v;
    __syncthreads();
  }
  const float prev = tid ? part[tid - 1] : 0.f;
  for (int i = 0; i < 16; ++i) cb[base + i] = loc[i] + prev;
}

// ---- K3: partial online logsumexp over t-chunks --------------------------
__global__ void __launch_bounds__(256) k3_partial_lse(
    const float* __restrict__ cb, const float* __restrict__ pcg,
    float* __restrict__ pm, float* __restrict__ ps)
{
  __shared__ float shcb[TCH_LEN];
  const int tid = threadIdx.x;
  const int j   = blockIdx.x * 256 + tid;   // candidate lane, coalesced
  const int cy  = blockIdx.y;
  int t0 = cy * TCH_LEN;
  const int t1 = t0 + TCH_LEN;
  // stage cb[t-1] for this chunk in LDS
  for (int i = tid; i < TCH_LEN; i += 256) {
    int t = cy * TCH_LEN + i;
    shcb[i] = (t >= 1) ? cb[t - 1] : NEG_INF_F;
  }
  __syncthreads();
  if (t0 < START_T) t0 = START_T;

  float m = NEG_INF_F, s = 0.f;
  for (int t = t0; t < t1; ++t) {
    __builtin_prefetch(&pcg[(size_t)(t + 8) * NB_DIM + j], 0, 0);
    float x = shcb[t - cy * TCH_LEN] + pcg[(size_t)t * NB_DIM + j];
    if (x > m) { s = s * __expf(m - x) + 1.f; m = x; }
    else       { s += __expf(x - m); }
  }
  pm[cy * NB_DIM + j] = m;
  ps[cy * NB_DIM + j] = s;
}

// ---- K4: combine partials, EOS override, write output --------------------
__global__ void __launch_bounds__(256) k4_combine(
    const float* __restrict__ pm, const float* __restrict__ ps,
    const float* __restrict__ cb, const int* __restrict__ c,
    float* __restrict__ out)
{
  const int j = blockIdx.x * 256 + threadIdx.x;
  if (j >= NB_DIM) return;
  float M = NEG_INF_F;
  for (int k = 0; k < TCHUNKS; ++k) M = fmaxf(M, pm[k * NB_DIM + j]);
  float score;
  if (M <= -1e29f) {
    score = NEG_INF_F;
  } else {
    float S = 0.f;
    for (int k = 0; k < TCHUNKS; ++k)
      S += ps[k * NB_DIM + j] * __expf(pm[k * NB_DIM + j] - M);
    score = M + __logf(S);
  }
  if (c[j] == EOS_ID) score = cb[T_DIM - 1];   // gamma_nb_g = logaddexp(cb[-1], -inf) = cb[-1]
  out[j] = score;
}

// ---- launch --------------------------------------------------------------
extern "C" void kernel_launch(void* const* d_in, const int* in_sizes, int n_in,
                              void* d_out, int out_size, void* d_ws, size_t ws_size,
                              hipStream_t stream)
{
  (void)in_sizes; (void)n_in; (void)out_size; (void)ws_size;
  const float* ctc_prob = (const float*)d_in[0];
  // d_in[1] = g is provably dead in the reference math (term == NEG_INF always).
  const int* c = (const int*)d_in[2];

  float* blank = (float*)d_ws;                      // T_DIM
  float* cb    = blank + T_DIM;                     // T_DIM
  float* pcg   = cb + T_DIM;                        // T_DIM * NB_DIM  (~33.5 MB)
  float* pm    = pcg + (size_t)T_DIM * NB_DIM;      // TCHUNKS * NB_DIM
  float* ps    = pm + TCHUNKS * NB_DIM;             // TCHUNKS * NB_DIM

  const size_t shmem = (size_t)(V_DIM + 64) * sizeof(float);  // ~128.3 KB of 320 KB/WGP
  k1_row_lse_gather<<<T_DIM, K1_THREADS, shmem, stream>>>(ctc_prob, c, blank, pcg);
  k2_cumsum<<<1, 256, 0, stream>>>(blank, cb);
  dim3 g3(NB_DIM / 256, TCHUNKS);
  k3_partial_lse<<<g3, 256, 0, stream>>>(cb, pcg, pm, ps);
  k4_combine<<<NB_DIM / 256, 256, 0, stream>>>(pm, ps, cb, c, (float*)d_out);
}